// RoiPooling_68805376082590
// MI455X (gfx1250) — compile-verified
//
#include <hip/hip_runtime.h>
#include <stdint.h>

#define NUM_ROIS 128
#define IMG_H    224
#define IMG_W    224
#define CHANS    128
#define BINS     21          // 1 + 4 + 16
#define CHUNK    4           // columns staged per async batch (512B per column)
#define NTHREADS 256         // 8 waves: (band 0..3) x (row parity 0..1)

// ---------------- CDNA5 async global->LDS copy path (ASYNCcnt) ----------------
#if defined(__gfx1250__)

typedef int v4i_t __attribute__((ext_vector_type(4)));

#if __has_builtin(__builtin_amdgcn_global_load_async_to_lds_b128)
__device__ __forceinline__ void async_cp_b128(const float* g, float* l) {
  // Builtin signature: void (global v4i*, local v4i*, imm offset, imm cpol)
  typedef __attribute__((address_space(1))) v4i_t* gp_t;   // global int4*
  typedef __attribute__((address_space(3))) v4i_t* lp_t;   // LDS int4*
  // low 32 bits of a flat shared address are the LDS byte offset
  __builtin_amdgcn_global_load_async_to_lds_b128(
      (gp_t)(uintptr_t)g,
      (lp_t)(uint32_t)(uintptr_t)l,
      0, 0);
}
#else
__device__ __forceinline__ void async_cp_b128(const float* g, float* l) {
  uint32_t loff = (uint32_t)(uintptr_t)l;   // LDS byte address
  asm volatile("global_load_async_to_lds_b128 %0, %1, off"
               :: "v"(loff), "v"(g) : "memory");
}
#endif

#if __has_builtin(__builtin_amdgcn_s_wait_asynccnt)
#define WAIT_ASYNC(n) do { __builtin_amdgcn_s_wait_asynccnt(n); \
                           asm volatile("" ::: "memory"); } while (0)
#else
#define WAIT_ASYNC(n) asm volatile("s_wait_asynccnt %0" :: "n"(n) : "memory")
#endif

#else  // host pass: never executed on device
__device__ __forceinline__ void async_cp_b128(const float*, float*) {}
#define WAIT_ASYNC(n) do {} while (0)
#endif

// Stage CHUNK columns [base, base+CHUNK) (clamped into [.., cend)) of one image
// row into an LDS buffer. Each lane copies its 16B (4 channels); 32 lanes cover
// one column's full 512B. Always issues exactly CHUNK async ops so the
// in-flight count is a compile-time constant for s_wait_asynccnt.
__device__ __forceinline__ void stage_chunk(const float* __restrict__ rowp,
                                            int base, int cend,
                                            float* __restrict__ buf, int c4) {
#pragma unroll
  for (int i = 0; i < CHUNK; ++i) {
    int col = base + i;
    col = (col < cend) ? col : (cend - 1);       // pad tail with last column
    async_cp_b128(rowp + (size_t)col * CHANS + c4, buf + i * CHANS + c4);
  }
}

__device__ __forceinline__ float4 max4(float4 a, float4 b) {
  a.x = fmaxf(a.x, b.x); a.y = fmaxf(a.y, b.y);
  a.z = fmaxf(a.z, b.z); a.w = fmaxf(a.w, b.w);
  return a;
}

__global__ __launch_bounds__(NTHREADS)
void roi_pyramid_pool_kernel(const float* __restrict__ img,
                             const float* __restrict__ rois,
                             float* __restrict__ out) {
  // 8 waves x 2 buffers x (CHUNK cols * 128 ch) = 32 KB staging
  __shared__ __align__(16) float stage[8][2][CHUNK * CHANS];
  // level-4 bin maxima [ix4*4+jy4][channel] = 8 KB
  __shared__ __align__(16) float red[16][CHANS];

  const int roi  = blockIdx.x;
  const int tid  = threadIdx.x;
  const int wave = tid >> 5;     // 0..7
  const int lane = tid & 31;
  const int band = wave >> 1;    // row band ix4 = 0..3
  const int par  = wave & 1;     // row parity within band
  const int c4   = lane * 4;     // this lane's channel base (float4)

  // ROI box; x indexes rows (H axis), y indexes cols (W axis), per reference.
  const float x = rois[roi * 4 + 0];
  const float y = rois[roi * 4 + 1];
  const float w = rois[roi * 4 + 2];
  const float h = rois[roi * 4 + 3];

  // Level-4 edges. Exact fp32 identities: 2*(w/4)==w/2, 4*(w/4)==w, so the
  // n=1 and n=2 partitions nest exactly inside the n=4 partition.
  int bx[5], by[5];
#pragma unroll
  for (int k = 0; k < 5; ++k) {
    bx[k] = (int)rintf(x + (float)k * (w * 0.25f));   // row edges
    by[k] = (int)rintf(y + (float)k * (h * 0.25f));   // col edges
  }

  float4 acc[4];
#pragma unroll
  for (int j = 0; j < 4; ++j)
    acc[j] = make_float4(-INFINITY, -INFINITY, -INFINITY, -INFINITY);

  float* buf0 = &stage[wave][0][0];
  float* buf1 = &stage[wave][1][0];

  const int r0 = bx[band], r1 = bx[band + 1];
  for (int r = r0 + par; r < r1; r += 2) {
    const float* rowp = img + (size_t)r * IMG_W * CHANS;
#pragma unroll
    for (int j = 0; j < 4; ++j) {               // jy4 band (static acc index)
      const int ca = by[j], cb = by[j + 1];
      if (ca >= cb) continue;
      stage_chunk(rowp, ca, cb, buf0, c4);      // prime the pipeline
      float4 a = acc[j];
      int parity = 0;
      for (int base = ca; base < cb; base += CHUNK) {
        const int nb = base + CHUNK;
        float* cur = parity ? buf1 : buf0;
        float* nxt = parity ? buf0 : buf1;
        if (nb < cb) {
          stage_chunk(rowp, nb, cb, nxt, c4);   // issue ahead
          WAIT_ASYNC(CHUNK);                    // prev CHUNK retired (in-order)
        } else {
          WAIT_ASYNC(0);                        // drain at band end
        }
        const int valid = (cb - base < CHUNK) ? (cb - base) : CHUNK;
        for (int i = 0; i < valid; ++i) {       // ds_load_b128 + 4x v_max
          float4 v = *(const float4*)(cur + i * CHANS + c4);
          a = max4(a, v);
        }
        parity ^= 1;
      }
      acc[j] = a;
    }
  }

  // Combine the row-parity wave pair per band into red[16][128].
  if (par == 1) {
#pragma unroll
    for (int j = 0; j < 4; ++j)
      *(float4*)&red[band * 4 + j][c4] = acc[j];
  }
  __syncthreads();
  if (par == 0) {
#pragma unroll
    for (int j = 0; j < 4; ++j) {
      float4 o = *(float4*)&red[band * 4 + j][c4];
      *(float4*)&red[band * 4 + j][c4] = max4(acc[j], o);
    }
  }
  __syncthreads();

  // Emit 21 bins x 128 channels, bin order: [L1 (1)], [L2 (ix2*2+jy2)],
  // [L4 (ix4*4+jy4)], channel-minor — matching the reference concat/reshape.
  float* o = out + (size_t)roi * (BINS * CHANS);
  for (int idx = tid; idx < BINS * CHANS; idx += NTHREADS) {
    const int bin = idx >> 7;
    const int c   = idx & (CHANS - 1);
    float v;
    if (bin >= 5) {                       // level 4: direct
      v = red[bin - 5][c];
    } else if (bin >= 1) {                // level 2: max of 2x2 level-4 bins
      const int b  = bin - 1;
      const int bb = (b >> 1) * 8 + (b & 1) * 2;  // (2*ix2)*4 + 2*jy2
      v = fmaxf(fmaxf(red[bb][c], red[bb + 1][c]),
                fmaxf(red[bb + 4][c], red[bb + 5][c]));
    } else {                              // level 1: max of all 16
      v = red[0][c];
#pragma unroll
      for (int b = 1; b < 16; ++b) v = fmaxf(v, red[b][c]);
    }
    o[idx] = v;
  }
}

extern "C" void kernel_launch(void* const* d_in, const int* in_sizes, int n_in,
                              void* d_out, int out_size, void* d_ws, size_t ws_size,
                              hipStream_t stream) {
  (void)in_sizes; (void)n_in; (void)d_ws; (void)ws_size; (void)out_size;
  const float* img  = (const float*)d_in[0];   // [1,224,224,128] f32
  const float* rois = (const float*)d_in[1];   // [1,128,4] f32
  float* out        = (float*)d_out;           // [1,128,21*128] f32
  hipLaunchKernelGGL(roi_pyramid_pool_kernel,
                     dim3(NUM_ROIS), dim3(NTHREADS), 0, stream,
                     img, rois, out);
}